// MeanAggregator_41214506173068
// MI455X (gfx1250) — compile-verified
//
#include <hip/hip_runtime.h>

// ---------------------------------------------------------------------------
// GCN mean aggregator for MI455X (gfx1250, wave32).
//   h   = x @ W^T + b          -> V_WMMA_F32_16X16X4_F32 (fp32 matrix pipe)
//   deg = segment_sum(1, src)  -> fused into scatter (lane 0 atomic)
//   agg = segment_sum(h[dst])  -> 1 wave/edge, b128 gather + atomic f32 adds
//   out = agg / deg[:,None]
// L2 (192MB) holds both h (51.2MB) and out (51.2MB), so the edge phase is
// L2-resident; atomics resolve at L2.
// ---------------------------------------------------------------------------

typedef __attribute__((ext_vector_type(2))) float v2f;
typedef __attribute__((ext_vector_type(8))) float v8f;
typedef __attribute__((ext_vector_type(4))) float f4v;

#define N_NODES 100000
#define N_EDGES 1600000
#define DFEAT   128

// -------------------------- zero out + deg ---------------------------------
__global__ void ma_init_kernel(float* __restrict__ out, float* __restrict__ deg) {
    long long i = (long long)blockIdx.x * blockDim.x + threadIdx.x;
    if (i < (long long)N_NODES * DFEAT) out[i] = 0.0f;
    if (i < N_NODES) deg[i] = 0.0f;
}

// -------------------------- h = x @ W^T + b --------------------------------
// Block = 256 threads = 8 waves; block b owns node rows [16b, 16b+16).
// Wave w owns output columns [16w, 16w+16). K loop: 32 x wmma_f32_16x16x4_f32.
//
// A (16x4 f32) lane layout (ISA 7.12.2): lanes 0-15 hold row M=lane with
// {VGPR0,VGPR1} = {K=0,K=1}; lanes 16-31 hold {K=2,K=3}. So each lane loads
// one float2 at x[(row0+idx)*128 + kk + 2*half].
// B (4x16 f32) mirrors with N striped across lanes: lane (half,idx) loads
// float2 at W^T[kk+2*half, o0+idx] == W[(o0+idx)*128 + kk + 2*half] — the
// row-major W row IS the W^T column, no transpose load needed.
__global__ __launch_bounds__(256)
void ma_gemm_kernel(const float* __restrict__ x, const float* __restrict__ W,
                    const float* __restrict__ bias, float* __restrict__ h) {
    const int wave = threadIdx.x >> 5;     // output tile 0..7
    const int lane = threadIdx.x & 31;
    const int half = lane >> 4;            // 0: lanes 0-15, 1: lanes 16-31
    const int idx  = lane & 15;
    const int row0 = blockIdx.x * 16;
    const int o0   = wave * 16;

    const float* xrow = x + (size_t)(row0 + idx) * DFEAT + 2 * half;
    const float* wrow = W + (size_t)(o0  + idx) * DFEAT + 2 * half;

    v8f c = {};
#pragma unroll
    for (int kk = 0; kk < DFEAT; kk += 4) {
        v2f a = *(const v2f*)(xrow + kk);
        v2f b = *(const v2f*)(wrow + kk);
        // 8 args: (neg_a, A, neg_b, B, c_mod, C, reuse_a, reuse_b)
        c = __builtin_amdgcn_wmma_f32_16x16x4_f32(
                false, a, false, b, (short)0, c, false, false);
    }

    const float bv = bias[o0 + idx];
    // C/D layout: VGPR r -> row M = r + 8*half, col N = idx.
    float* hout = h + (size_t)row0 * DFEAT + o0 + idx;
#pragma unroll
    for (int r = 0; r < 8; ++r) {
        hout[(size_t)(r + 8 * half) * DFEAT] = c[r] + bv;
    }
}

// ---------------- agg[src] += h[dst];  deg[src] += 1 -----------------------
// One wave per edge; lane covers 4 consecutive floats (b128 gather).
__global__ __launch_bounds__(256)
void ma_scatter_kernel(const int* __restrict__ esrc, const int* __restrict__ edst,
                       const float* __restrict__ h, float* __restrict__ out,
                       float* __restrict__ deg) {
    const int lane = threadIdx.x & 31;
    const long long e = (long long)blockIdx.x * 8 + (threadIdx.x >> 5);
    if (e >= N_EDGES) return;
    const int s = esrc[e];
    const int d = edst[e];

    f4v v = *(const f4v*)(h + (size_t)d * DFEAT + lane * 4);
    float* o = out + (size_t)s * DFEAT + lane * 4;
    atomicAdd(o + 0, v.x);
    atomicAdd(o + 1, v.y);
    atomicAdd(o + 2, v.z);
    atomicAdd(o + 3, v.w);
    if (lane == 0) atomicAdd(deg + s, 1.0f);
}

// -------------------------- out /= deg -------------------------------------
__global__ void ma_divide_kernel(float* __restrict__ out, const float* __restrict__ deg) {
    long long i = (long long)blockIdx.x * blockDim.x + threadIdx.x;
    if (i < (long long)N_NODES * DFEAT) {
        out[i] = out[i] / deg[i >> 7];   // DFEAT == 128
    }
}

// ---------------------------------------------------------------------------
extern "C" void kernel_launch(void* const* d_in, const int* in_sizes, int n_in,
                              void* d_out, int out_size, void* d_ws, size_t ws_size,
                              hipStream_t stream) {
    const float* x    = (const float*)d_in[0];
    const int*   esrc = (const int*)  d_in[1];
    const int*   edst = (const int*)  d_in[2];
    const float* W    = (const float*)d_in[3];
    const float* bias = (const float*)d_in[4];
    float* out = (float*)d_out;

    // workspace layout: h [N_NODES*DFEAT] floats, then deg [N_NODES] floats
    float* h   = (float*)d_ws;
    float* deg = h + (size_t)N_NODES * DFEAT;

    const long long nd = (long long)N_NODES * DFEAT;

    ma_init_kernel   <<<(unsigned)((nd + 255) / 256), 256, 0, stream>>>(out, deg);
    ma_gemm_kernel   <<<N_NODES / 16, 256, 0, stream>>>(x, W, bias, h);
    ma_scatter_kernel<<<(N_EDGES + 7) / 8, 256, 0, stream>>>(esrc, edst, h, out, deg);
    ma_divide_kernel <<<(unsigned)((nd + 255) / 256), 256, 0, stream>>>(out, deg);
}